// BlockBottleneckedInterconnect_59425167507868
// MI455X (gfx1250) — compile-verified
//
#include <hip/hip_runtime.h>
#include <hip/hip_bf16.h>

#define N_BLOCKS  512
#define BLOCK_IN  128
#define BLOCK_OUT 128
#define BATCH     2048
#define LAYER_IN  (N_BLOCKS * BLOCK_IN)   // 65536
#define LAYER_OUT (N_BLOCKS * BLOCK_OUT)  // 65536
#define BTILE     64
#define XPAD_STRIDE (BLOCK_IN + 4)        // 132 floats: conflict-free A reads (528B rows, 16B aligned)
#define KPAIRS    (BLOCK_IN / 2)          // 64 pair-rows
#define WROW      (BLOCK_OUT * 2)         // 256 floats per pair-row in memory
#define WROW_LDS  (WROW + 32)             // 288 floats (1152B): odd/even pair-rows hit disjoint bank halves

typedef __attribute__((ext_vector_type(2))) float v2f;
typedef __attribute__((ext_vector_type(8))) float v8f;
typedef __attribute__((ext_vector_type(4))) int   v4i;

// ---- async global->LDS path (CDNA5 GLOBAL_LOAD_ASYNC_TO_LDS_B128, ASYNCcnt) ----
#if __has_builtin(__builtin_amdgcn_global_load_async_to_lds_b128) && \
    __has_builtin(__builtin_amdgcn_s_wait_asynccnt)
#define USE_ASYNC_LDS 1
typedef __attribute__((address_space(1))) v4i gbl_v4i;  // global int4
typedef __attribute__((address_space(3))) v4i lds_v4i;  // LDS int4
__device__ __forceinline__ void async_cp16(const void* g, void* l) {
  __builtin_amdgcn_global_load_async_to_lds_b128((gbl_v4i*)g, (lds_v4i*)l, 0, 0);
}
__device__ __forceinline__ void async_wait0() {
  __builtin_amdgcn_s_wait_asynccnt(0);
}
#else
#define USE_ASYNC_LDS 0
#endif

// ---------------------------------------------------------------------------
// Kernel 1: softmax over m of c[n][m][o], written PAIR-INTERLEAVED:
//   wp[n][kp][o][j] = softmax_m(c[n])[2*kp+j][o]
// so a WMMA-f32 B fragment {W[k][o], W[k+1][o]} is 8 contiguous bytes.
// ---------------------------------------------------------------------------
__global__ __launch_bounds__(128) void softmax_k_kernel(
    const float* __restrict__ c, float* __restrict__ wp) {
  __shared__ float tile[BLOCK_IN * BLOCK_OUT]; // 64 KiB
  const int n = blockIdx.x;
  const float* cn = c  + (size_t)n * BLOCK_IN * BLOCK_OUT;
  float*       wn = wp + (size_t)n * KPAIRS * WROW;

  const float4* c4 = (const float4*)cn;
  float4*       t4 = (float4*)tile;
  for (int i = threadIdx.x; i < BLOCK_IN * BLOCK_OUT / 4; i += 128)
    t4[i] = c4[i];
  __syncthreads();

  const int o = threadIdx.x; // one output column per thread
  float mx = -INFINITY;
  #pragma unroll 8
  for (int m = 0; m < BLOCK_IN; ++m)
    mx = fmaxf(mx, tile[m * BLOCK_OUT + o]);
  float s = 0.f;
  #pragma unroll 8
  for (int m = 0; m < BLOCK_IN; ++m) {
    float e = __expf(tile[m * BLOCK_OUT + o] - mx);
    tile[m * BLOCK_OUT + o] = e;
    s += e;
  }
  const float r = 1.f / s;
  #pragma unroll 8
  for (int kp = 0; kp < KPAIRS; ++kp) {
    float2 v;
    v.x = tile[(2 * kp)     * BLOCK_OUT + o] * r;
    v.y = tile[(2 * kp + 1) * BLOCK_OUT + o] * r;
    *(float2*)&wn[kp * WROW + 2 * o] = v;   // coalesced 8B/lane stores
  }
}

// ---------------------------------------------------------------------------
// Kernel 2: block-diagonal GEMM with V_WMMA_F32_16X16X4_F32.
// grid = (512 blocks, 32 batch-tiles), 256 threads = 8 wave32.
// Wave w: M-tile = w%4 (16 rows), N-half = w/4 (64 cols -> 4 WMMA N-tiles).
// ---------------------------------------------------------------------------
__global__ __launch_bounds__(256) void block_gemm_wmma_kernel(
    const float* __restrict__ x, const float* __restrict__ wp,
    float* __restrict__ out) {
  __shared__ float Wl[KPAIRS * WROW_LDS];   // 72 KiB, pair-interleaved + padded
  __shared__ float Xl[BTILE * XPAD_STRIDE]; // 33 KiB, [row][k] padded

  const int n    = blockIdx.x;   // block id 0..511
  const int bt   = blockIdx.y;   // batch tile 0..31
  const int tid  = threadIdx.x;
  const int lane = tid & 31;
  const int wv   = tid >> 5;     // wave 0..7
  const int mt   = wv & 3;       // M-tile within batch tile
  const int nh   = wv >> 2;      // N-half (0 or 1)

  const char* wsrc = (const char*)(wp + (size_t)n * KPAIRS * WROW);
  const float* xb  = x + ((size_t)bt * BTILE) * LAYER_IN + (size_t)n * BLOCK_IN;

#if USE_ASYNC_LDS
  // ---- async DMA staging: global -> LDS directly (no VGPR round trip) ----
  #pragma unroll 4
  for (int i = tid; i < KPAIRS * (WROW / 4); i += 256) {
    const int kp = i >> 6;       // / (256/4)
    const int q  = i & 63;
    async_cp16(wsrc + (size_t)i * 16, (char*)Wl + kp * (WROW_LDS * 4) + q * 16);
  }
  #pragma unroll 4
  for (int i = tid; i < BTILE * (BLOCK_IN / 4); i += 256) {
    const int row = i >> 5;
    const int q   = i & 31;
    async_cp16((const char*)(xb + (size_t)row * LAYER_IN) + q * 16,
               (char*)Xl + row * (XPAD_STRIDE * 4) + q * 16);
  }
  async_wait0();      // this wave's async copies landed in LDS
  __syncthreads();    // all waves' copies visible
#else
  {
    const float4* src = (const float4*)wsrc;
    #pragma unroll 4
    for (int i = tid; i < KPAIRS * (WROW / 4); i += 256) {
      const int kp = i >> 6;
      const int q  = i & 63;
      *(float4*)&Wl[kp * WROW_LDS + q * 4] = src[i];
    }
  }
  {
    #pragma unroll 4
    for (int i = tid; i < BTILE * (BLOCK_IN / 4); i += 256) {
      const int row = i >> 5;
      const int q   = i & 31;
      float4 v = *(const float4*)(xb + (size_t)row * LAYER_IN + q * 4);
      float* d = &Xl[row * XPAD_STRIDE + q * 4];
      d[0] = v.x; d[1] = v.y; d[2] = v.z; d[3] = v.w;
    }
  }
  __syncthreads();
#endif

  v8f acc0 = {}, acc1 = {}, acc2 = {}, acc3 = {};

  // A frag (16x4 f32): lanes 0-15 -> M=lane, K={k0,k0+1}; lanes 16-31 -> K={k0+2,k0+3}
  const int arow  = mt * 16 + (lane & 15);
  const int khalf = (lane >> 4) * 2;             // 0 or 2
  const int bcol2 = (nh * 64 + (lane & 15)) * 2; // pair-interleaved column offset

  #pragma unroll 4
  for (int k0 = 0; k0 < BLOCK_IN; k0 += 4) {
    const v2f a = *(const v2f*)&Xl[arow * XPAD_STRIDE + k0 + khalf];
    const float* wrow = &Wl[((k0 + khalf) >> 1) * WROW_LDS + bcol2];
    const v2f b0 = *(const v2f*)(wrow);        // {W[k][o],    W[k+1][o]}
    const v2f b1 = *(const v2f*)(wrow + 32);   // o + 16
    const v2f b2 = *(const v2f*)(wrow + 64);   // o + 32
    const v2f b3 = *(const v2f*)(wrow + 96);   // o + 48
    acc0 = __builtin_amdgcn_wmma_f32_16x16x4_f32(false, a, false, b0, (short)0, acc0, false, false);
    acc1 = __builtin_amdgcn_wmma_f32_16x16x4_f32(false, a, false, b1, (short)0, acc1, false, false);
    acc2 = __builtin_amdgcn_wmma_f32_16x16x4_f32(false, a, false, b2, (short)0, acc2, false, false);
    acc3 = __builtin_amdgcn_wmma_f32_16x16x4_f32(false, a, false, b3, (short)0, acc3, false, false);
  }

  // ---- store C/D: VGPR g -> M = g + (lane<16 ? 0 : 8), N = lane&15 ----
  float* outn = out + (size_t)n * BLOCK_OUT;
  const int mbase = bt * BTILE + mt * 16 + ((lane >> 4) * 8);
  const int nc0   = nh * 64 + (lane & 15);
  #pragma unroll
  for (int g = 0; g < 8; ++g) {
    const size_t rowoff = (size_t)(mbase + g) * LAYER_OUT;
    outn[rowoff + nc0     ] = acc0[g];
    outn[rowoff + nc0 + 16] = acc1[g];
    outn[rowoff + nc0 + 32] = acc2[g];
    outn[rowoff + nc0 + 48] = acc3[g];
  }
}

// ---------------------------------------------------------------------------
extern "C" void kernel_launch(void* const* d_in, const int* in_sizes, int n_in,
                              void* d_out, int out_size, void* d_ws, size_t ws_size,
                              hipStream_t stream) {
  (void)in_sizes; (void)n_in; (void)out_size; (void)ws_size;
  const float* x = (const float*)d_in[0];   // [2048, 65536]
  const float* c = (const float*)d_in[1];   // [512, 128, 128]
  float* out = (float*)d_out;               // [2048, 65536]
  float* wp  = (float*)d_ws;                // 32 MiB pair-interleaved softmax(c)

  softmax_k_kernel<<<dim3(N_BLOCKS), dim3(128), 0, stream>>>(c, wp);
  block_gemm_wmma_kernel<<<dim3(N_BLOCKS, BATCH / BTILE), dim3(256), 0, stream>>>(x, wp, out);
}